// FrequencyBranch_86011015070197
// MI455X (gfx1250) — compile-verified
//
#include <hip/hip_runtime.h>
#include <math.h>

typedef __attribute__((ext_vector_type(2))) float v2f;
typedef __attribute__((ext_vector_type(8))) float v8f;

#define DN 224
#define DNN (DN * DN)
#define DHALF 112
#define NIMG 48               // 16 batches * 3 channels
#define GRP 7                 // 224 / 32 : 2x2-tile groups per dimension
#define GRP_PER_IMG (GRP * GRP)
#define GEMM_WAVES (NIMG * GRP_PER_IMG)   // 2352

#define WMMA_F32(A, B, C) \
  __builtin_amdgcn_wmma_f32_16x16x4_f32(false, (A), false, (B), (short)0, (C), false, false)

// ---------------------------------------------------------------------------
// Stage 1: Cx = C @ x, Sx = S @ x  for 48 images. fp32 WMMA 16x16x4 with
// 2x2 output-tile register blocking: each wave computes a 32x32 block of
// BOTH outputs -> 8 WMMAs per k-step from 4 A-fragments + 2 B-fragments.
// ---------------------------------------------------------------------------
__global__ __launch_bounds__(256) void dft_stage1(
    const float* __restrict__ seq, const float* __restrict__ Cm,
    const float* __restrict__ Sm, float* __restrict__ Cx,
    float* __restrict__ Sx)
{
  int wave = blockIdx.x * (blockDim.x >> 5) + (threadIdx.x >> 5);
  if (wave >= GEMM_WAVES) return;           // wave-uniform (EXEC stays all-1)
  int lane = threadIdx.x & 31;
  int img  = wave / GRP_PER_IMG;
  int t    = wave % GRP_PER_IMG;
  int gm = t / GRP, gn = t % GRP;
  int b = img / 3, c = img % 3;
  const float* x = seq + (((size_t)(b * 8 + 4)) * 3 + (size_t)c) * DNN;

  int half = lane >> 4;                     // lane-half selects K pair
  int l16  = lane & 15;
  int row0 = gm * 32 + l16, row1 = row0 + 16;
  int col0 = gn * 32 + l16, col1 = col0 + 16;

  v8f aC00 = {}, aC01 = {}, aC10 = {}, aC11 = {};
  v8f aS00 = {}, aS01 = {}, aS10 = {}, aS11 = {};

  for (int k0 = 0; k0 < DN; k0 += 4) {
    int ka = k0 + 2 * half;
    v2f A0c = *(const v2f*)(Cm + row0 * DN + ka);
    v2f A1c = *(const v2f*)(Cm + row1 * DN + ka);
    v2f A0s = *(const v2f*)(Sm + row0 * DN + ka);
    v2f A1s = *(const v2f*)(Sm + row1 * DN + ka);
    v2f B0, B1;
    B0.x = x[ka * DN + col0];  B0.y = x[(ka + 1) * DN + col0];
    B1.x = x[ka * DN + col1];  B1.y = x[(ka + 1) * DN + col1];

    aC00 = WMMA_F32(A0c, B0, aC00);  aC01 = WMMA_F32(A0c, B1, aC01);
    aC10 = WMMA_F32(A1c, B0, aC10);  aC11 = WMMA_F32(A1c, B1, aC11);
    aS00 = WMMA_F32(A0s, B0, aS00);  aS01 = WMMA_F32(A0s, B1, aS01);
    aS10 = WMMA_F32(A1s, B0, aS10);  aS11 = WMMA_F32(A1s, B1, aS11);
  }

  float* oC = Cx + (size_t)img * DNN;
  float* oS = Sx + (size_t)img * DNN;
#pragma unroll
  for (int v = 0; v < 8; ++v) {
    int m0 = gm * 32 + v + 8 * half;        // C/D layout: M = v + 8*half
    int m1 = m0 + 16;
    oC[m0 * DN + col0] = aC00[v];  oC[m0 * DN + col1] = aC01[v];
    oC[m1 * DN + col0] = aC10[v];  oC[m1 * DN + col1] = aC11[v];
    oS[m0 * DN + col0] = aS00[v];  oS[m0 * DN + col1] = aS01[v];
    oS[m1 * DN + col0] = aS10[v];  oS[m1 * DN + col1] = aS11[v];
  }
}

// ---------------------------------------------------------------------------
// Stage 2 epilogue helper: combine 4 accumulators into log-magnitude and
// store with roll(112,112) applied.
// ---------------------------------------------------------------------------
__device__ __forceinline__ void store_spec(float* __restrict__ o, int mbase,
                                           int col, int half, const v8f& rc,
                                           const v8f& rs, const v8f& ic,
                                           const v8f& is_)
{
#pragma unroll
  for (int v = 0; v < 8; ++v) {
    int m = mbase + v + 8 * half;
    float xr = rc[v] - rs[v];
    float xi = ic[v] + is_[v];              // sign irrelevant under square
    float mag = sqrtf(xr * xr + xi * xi + 1e-8f);
    float sp  = logf(mag + 1e-8f);
    int mi = (m + DHALF) % DN;              // out[(i+112)%224][(j+112)%224] = in[i][j]
    int mj = (col + DHALF) % DN;
    o[mi * DN + mj] = sp;
  }
}

// ---------------------------------------------------------------------------
// Stage 2: Xr = E@C - F@S, Xi = E@S + F@C (sign dropped), fused |.|, log,
// roll. 2x2 register blocking: 16 WMMAs per k-step from 4 A + 4 B fragments.
// ---------------------------------------------------------------------------
__global__ __launch_bounds__(256) void dft_stage2(
    const float* __restrict__ Cx, const float* __restrict__ Sx,
    const float* __restrict__ Cm, const float* __restrict__ Sm,
    float* __restrict__ spec)
{
  int wave = blockIdx.x * (blockDim.x >> 5) + (threadIdx.x >> 5);
  if (wave >= GEMM_WAVES) return;
  int lane = threadIdx.x & 31;
  int img  = wave / GRP_PER_IMG;
  int t    = wave % GRP_PER_IMG;
  int gm = t / GRP, gn = t % GRP;
  const float* E = Cx + (size_t)img * DNN;
  const float* F = Sx + (size_t)img * DNN;

  int half = lane >> 4;
  int l16  = lane & 15;
  int row0 = gm * 32 + l16, row1 = row0 + 16;
  int col0 = gn * 32 + l16, col1 = col0 + 16;

  // RC = E@C, RS = F@S, IC = E@S, IS = F@C, each over a 2x2 tile block.
  v8f rc00 = {}, rc01 = {}, rc10 = {}, rc11 = {};
  v8f rs00 = {}, rs01 = {}, rs10 = {}, rs11 = {};
  v8f ic00 = {}, ic01 = {}, ic10 = {}, ic11 = {};
  v8f is00 = {}, is01 = {}, is10 = {}, is11 = {};

  for (int k0 = 0; k0 < DN; k0 += 4) {
    int ka = k0 + 2 * half;
    v2f aE0 = *(const v2f*)(E + row0 * DN + ka);
    v2f aE1 = *(const v2f*)(E + row1 * DN + ka);
    v2f aF0 = *(const v2f*)(F + row0 * DN + ka);
    v2f aF1 = *(const v2f*)(F + row1 * DN + ka);
    v2f bC0, bC1, bS0, bS1;
    bC0.x = Cm[ka * DN + col0];  bC0.y = Cm[(ka + 1) * DN + col0];
    bC1.x = Cm[ka * DN + col1];  bC1.y = Cm[(ka + 1) * DN + col1];
    bS0.x = Sm[ka * DN + col0];  bS0.y = Sm[(ka + 1) * DN + col0];
    bS1.x = Sm[ka * DN + col1];  bS1.y = Sm[(ka + 1) * DN + col1];

    rc00 = WMMA_F32(aE0, bC0, rc00);  rc01 = WMMA_F32(aE0, bC1, rc01);
    rc10 = WMMA_F32(aE1, bC0, rc10);  rc11 = WMMA_F32(aE1, bC1, rc11);
    rs00 = WMMA_F32(aF0, bS0, rs00);  rs01 = WMMA_F32(aF0, bS1, rs01);
    rs10 = WMMA_F32(aF1, bS0, rs10);  rs11 = WMMA_F32(aF1, bS1, rs11);
    ic00 = WMMA_F32(aE0, bS0, ic00);  ic01 = WMMA_F32(aE0, bS1, ic01);
    ic10 = WMMA_F32(aE1, bS0, ic10);  ic11 = WMMA_F32(aE1, bS1, ic11);
    is00 = WMMA_F32(aF0, bC0, is00);  is01 = WMMA_F32(aF0, bC1, is01);
    is10 = WMMA_F32(aF1, bC0, is10);  is11 = WMMA_F32(aF1, bC1, is11);
  }

  float* o = spec + (size_t)img * DNN;
  int m0 = gm * 32, m1 = m0 + 16;
  store_spec(o, m0, col0, half, rc00, rs00, ic00, is00);
  store_spec(o, m0, col1, half, rc01, rs01, ic01, is01);
  store_spec(o, m1, col0, half, rc10, rs10, ic10, is10);
  store_spec(o, m1, col1, half, rc11, rs11, ic11, is11);
}

// ---------------------------------------------------------------------------
// Generic 3x3 SAME conv + BN + ReLU, optionally fused 2x2 maxpool.
// ---------------------------------------------------------------------------
__global__ __launch_bounds__(256) void conv_bn_relu(
    const float* __restrict__ in, const float* __restrict__ w,
    const float* __restrict__ bias, const float* __restrict__ g,
    const float* __restrict__ beta, const float* __restrict__ mean,
    const float* __restrict__ var, float* __restrict__ out,
    int B, int Cin, int Cout, int H, int pool)
{
  int Hout = pool ? (H >> 1) : H;
  int total = B * Cout * Hout * Hout;
  int idx = blockIdx.x * blockDim.x + threadIdx.x;
  if (idx >= total) return;
  int qo = idx % Hout; int tt = idx / Hout;
  int po = tt % Hout;  tt /= Hout;
  int oc = tt % Cout;  int b = tt / Cout;

  float scale = g[oc] * rsqrtf(var[oc] + 1e-5f);
  float shift = beta[oc] + (bias[oc] - mean[oc]) * scale;

  int span = pool ? 2 : 1;
  float best = 0.0f;                        // ReLU outputs are >= 0
  for (int di = 0; di < span; ++di) {
    for (int dj = 0; dj < span; ++dj) {
      int oi = po * span + di;
      int oj = qo * span + dj;
      float s = 0.0f;
      for (int ic = 0; ic < Cin; ++ic) {
        const float* xp = in + ((size_t)(b * Cin + ic)) * (size_t)(H * H);
        const float* wp = w + ((size_t)(oc * Cin + ic)) * 9;
#pragma unroll
        for (int ki = 0; ki < 3; ++ki) {
          int ii = oi + ki - 1;
          if ((unsigned)ii >= (unsigned)H) continue;
#pragma unroll
          for (int kj = 0; kj < 3; ++kj) {
            int jj = oj + kj - 1;
            if ((unsigned)jj >= (unsigned)H) continue;
            s = fmaf(xp[ii * H + jj], wp[ki * 3 + kj], s);
          }
        }
      }
      float y = fmaxf(fmaf(s, scale, shift), 0.0f);
      best = fmaxf(best, y);
    }
  }
  out[idx] = best;
}

// ---------------------------------------------------------------------------
// Mean over H*W per (batch, channel). One 256-thread block per (b, c).
// ---------------------------------------------------------------------------
__global__ __launch_bounds__(256) void mean_hw(const float* __restrict__ in,
                                               float* __restrict__ out, int HW)
{
  __shared__ float sm[256];
  int bc = blockIdx.x;
  const float* p = in + (size_t)bc * HW;
  float s = 0.0f;
  for (int i = threadIdx.x; i < HW; i += blockDim.x) s += p[i];
  sm[threadIdx.x] = s;
  __syncthreads();
  for (int stride = 128; stride > 0; stride >>= 1) {
    if ((int)threadIdx.x < stride) sm[threadIdx.x] += sm[threadIdx.x + stride];
    __syncthreads();
  }
  if (threadIdx.x == 0) out[bc] = sm[0] / (float)HW;
}

// ---------------------------------------------------------------------------
// FC: out[b][o] = relu(sum_k h[b][k] * fc_w[o][k] + fc_b[o]),  (16,64)->(16,128)
// ---------------------------------------------------------------------------
__global__ __launch_bounds__(256) void fc_relu(const float* __restrict__ hm,
                                               const float* __restrict__ w,
                                               const float* __restrict__ bv,
                                               float* __restrict__ out)
{
  int idx = blockIdx.x * blockDim.x + threadIdx.x;
  if (idx >= 16 * 128) return;
  int o = idx % 128, b = idx / 128;
  float s = bv[o];
  const float* hp = hm + b * 64;
  const float* wp = w + o * 64;
#pragma unroll
  for (int k = 0; k < 64; ++k) s = fmaf(hp[k], wp[k], s);
  out[idx] = fmaxf(s, 0.0f);
}

// ---------------------------------------------------------------------------
extern "C" void kernel_launch(void* const* d_in, const int* in_sizes, int n_in,
                              void* d_out, int out_size, void* d_ws,
                              size_t ws_size, hipStream_t stream)
{
  const float* seq  = (const float*)d_in[0];
  const float* dftC = (const float*)d_in[1];
  const float* dftS = (const float*)d_in[2];
  const float* c1w  = (const float*)d_in[3];
  const float* c1b  = (const float*)d_in[4];
  const float* b1g  = (const float*)d_in[5];
  const float* b1b  = (const float*)d_in[6];
  const float* b1m  = (const float*)d_in[7];
  const float* b1v  = (const float*)d_in[8];
  const float* c2w  = (const float*)d_in[9];
  const float* c2b  = (const float*)d_in[10];
  const float* b2g  = (const float*)d_in[11];
  const float* b2b  = (const float*)d_in[12];
  const float* b2m  = (const float*)d_in[13];
  const float* b2v  = (const float*)d_in[14];
  const float* c3w  = (const float*)d_in[15];
  const float* c3b  = (const float*)d_in[16];
  const float* b3g  = (const float*)d_in[17];
  const float* b3b  = (const float*)d_in[18];
  const float* b3m  = (const float*)d_in[19];
  const float* b3v  = (const float*)d_in[20];
  const float* fcw  = (const float*)d_in[21];
  const float* fcb  = (const float*)d_in[22];

  float* ws   = (float*)d_ws;
  float* Cx   = ws;                                     // 48*224*224
  float* Sx   = Cx + (size_t)NIMG * DNN;                // 48*224*224
  float* spec = Sx + (size_t)NIMG * DNN;                // 48*224*224
  float* h1   = spec + (size_t)NIMG * DNN;              // 16*16*112*112
  float* h2   = h1 + (size_t)16 * 16 * 112 * 112;       // 16*32*56*56
  float* h3   = h2 + (size_t)16 * 32 * 56 * 56;         // 16*64*56*56
  float* hm   = h3 + (size_t)16 * 64 * 56 * 56;         // 16*64

  dim3 blk(256);
  int gemm_blocks = (GEMM_WAVES + 7) / 8;               // 8 waves per block

  dft_stage1<<<gemm_blocks, blk, 0, stream>>>(seq, dftC, dftS, Cx, Sx);
  dft_stage2<<<gemm_blocks, blk, 0, stream>>>(Cx, Sx, dftC, dftS, spec);

  int t1 = 16 * 16 * 112 * 112;
  conv_bn_relu<<<(t1 + 255) / 256, blk, 0, stream>>>(
      spec, c1w, c1b, b1g, b1b, b1m, b1v, h1, 16, 3, 16, 224, 1);

  int t2 = 16 * 32 * 56 * 56;
  conv_bn_relu<<<(t2 + 255) / 256, blk, 0, stream>>>(
      h1, c2w, c2b, b2g, b2b, b2m, b2v, h2, 16, 16, 32, 112, 1);

  int t3 = 16 * 64 * 56 * 56;
  conv_bn_relu<<<(t3 + 255) / 256, blk, 0, stream>>>(
      h2, c3w, c3b, b3g, b3b, b3m, b3v, h3, 16, 32, 64, 56, 0);

  mean_hw<<<16 * 64, blk, 0, stream>>>(h3, hm, 56 * 56);

  fc_relu<<<(16 * 128 + 255) / 256, blk, 0, stream>>>(hm, fcw, fcb,
                                                      (float*)d_out);
}